// appnp_53592601919957
// MI455X (gfx1250) — compile-verified
//
#include <hip/hip_runtime.h>
#include <hip/hip_bf16.h>

// ---------------------------------------------------------------------------
// APPNP for MI455X (gfx1250, wave32).
//   h0 = relu(x @ W + b)                           -> fp32 WMMA 16x16x4
//   CSR build (once per call): edges grouped by dst, packed {src, w} records,
//     w = 0.9*dinv[s]*dinv[d]
//   50x: h'[i] = alpha*x0[i] + 0.9*deg^-1[i]*h[i] + sum_in(i) w*h[src]
//        (pure gather SpMM, zero float atomics, L2-resident working set)
//   out = log_softmax(h, axis=1)
// ---------------------------------------------------------------------------

typedef __attribute__((ext_vector_type(2))) float v2f;
typedef __attribute__((ext_vector_type(8))) float v8f;

#define FIN 512
#define CCH 64
#define ALPHA 0.1f
#define BETA 0.9f
#define KSTEPS 50
#define SCAN_B 256   // elements per scan block
#define SCAN_T2 512  // threads in second-level scan (must cover #blocks)

// ---------------- degree -----------------------------------------------------
__global__ void k_deg_init(int* __restrict__ deg, int n) {
    int i = blockIdx.x * blockDim.x + threadIdx.x;
    if (i < n) deg[i] = 1;  // self-loop
}

__global__ void k_deg_count(int* __restrict__ deg, const int* __restrict__ dst, int e) {
    int i = blockIdx.x * blockDim.x + threadIdx.x;
    int stride = gridDim.x * blockDim.x;
    for (; i < e; i += stride) atomicAdd(&deg[dst[i]], 1);
}

__global__ void k_dinv(float* __restrict__ dinv, float* __restrict__ d2s,
                       const int* __restrict__ deg, int n) {
    int i = blockIdx.x * blockDim.x + threadIdx.x;
    if (i < n) {
        float r = rsqrtf((float)deg[i]);
        dinv[i] = r;
        d2s[i] = BETA * r * r;  // (1-alpha) * self-loop norm
    }
}

// ---------------- exclusive prefix scan of (deg[i]-1) -> rowstart ------------
__global__ __launch_bounds__(SCAN_B) void k_scan1(const int* __restrict__ deg,
                                                  int* __restrict__ rs,
                                                  int* __restrict__ partial, int n) {
    __shared__ int sm[SCAN_B];
    const int t = threadIdx.x;
    const int i = blockIdx.x * SCAN_B + t;
    const int v = (i < n) ? (deg[i] - 1) : 0;
    sm[t] = v;
    __syncthreads();
    for (int off = 1; off < SCAN_B; off <<= 1) {
        int add = (t >= off) ? sm[t - off] : 0;
        __syncthreads();
        sm[t] += add;
        __syncthreads();
    }
    if (i < n) rs[i] = sm[t] - v;  // exclusive
    if (t == SCAN_B - 1) partial[blockIdx.x] = sm[t];
}

__global__ __launch_bounds__(SCAN_T2) void k_scan2(int* __restrict__ partial, int nb) {
    __shared__ int sm[SCAN_T2];
    const int t = threadIdx.x;
    const int v = (t < nb) ? partial[t] : 0;
    sm[t] = v;
    __syncthreads();
    for (int off = 1; off < SCAN_T2; off <<= 1) {
        int add = (t >= off) ? sm[t - off] : 0;
        __syncthreads();
        sm[t] += add;
        __syncthreads();
    }
    if (t < nb) partial[t] = sm[t] - v;  // exclusive
}

__global__ void k_scan3(int* __restrict__ rs, int* __restrict__ pos,
                        const int* __restrict__ partial, int n) {
    int i = blockIdx.x * blockDim.x + threadIdx.x;
    if (i < n) {
        int v = rs[i] + partial[i / SCAN_B];
        rs[i] = v;
        pos[i] = v;  // running fill cursor for CSR build
    }
}

// ---------------- CSR fill (by dst): packed {src, weight_bits} records -------
__global__ void k_csr_fill(int* __restrict__ pos, int2* __restrict__ cedge,
                           const int* __restrict__ src, const int* __restrict__ dst,
                           const float* __restrict__ dinv, int e) {
    int i = blockIdx.x * blockDim.x + threadIdx.x;
    int stride = gridDim.x * blockDim.x;
    for (; i < e; i += stride) {
        const int s = src[i], d = dst[i];
        const int p = atomicAdd(&pos[d], 1);
        int2 rec;
        rec.x = s;
        rec.y = __float_as_int(BETA * dinv[s] * dinv[d]);
        cedge[p] = rec;
    }
}

// ---------------- GEMM + bias + relu (WMMA fp32 16x16x4) ---------------------
// Block = 256 threads = 8 waves; wave w owns rows [blk*128 + 16w, +16) x all 64 cols.
__global__ __launch_bounds__(256) void k_gemm_relu(const float* __restrict__ x,
                                                   const float* __restrict__ W,
                                                   const float* __restrict__ bias,
                                                   float* __restrict__ h0, int n) {
#if defined(__HIP_DEVICE_COMPILE__) && __has_builtin(__builtin_amdgcn_wmma_f32_16x16x4_f32)
    const int lane = threadIdx.x & 31;
    const int wave = threadIdx.x >> 5;
    const int row0 = blockIdx.x * 128 + wave * 16;
    if (row0 >= n) return;  // wave-uniform (n is a multiple of 16)

    const int halfSel = lane >> 4;  // 0: lanes 0-15, 1: lanes 16-31
    const int l15 = lane & 15;

    v8f z = {0.f, 0.f, 0.f, 0.f, 0.f, 0.f, 0.f, 0.f};
    v8f acc0 = z, acc1 = z, acc2 = z, acc3 = z;

    // A fragment: lane (l15,halfSel) holds x[row0+l15][k + 2*halfSel + {0,1}]
    const float* xrow = x + (size_t)(row0 + l15) * FIN + halfSel * 2;

    for (int k = 0; k < FIN; k += 4) {
        v2f a = *(const v2f*)(xrow + k);
        // B fragment for col tile t: VGPR r holds W[k + 2*halfSel + r][t*16 + l15]
        const float* wp = W + (size_t)(k + halfSel * 2) * CCH + l15;
        v2f b0, b1, b2, b3;
        b0.x = wp[0];      b0.y = wp[CCH + 0];
        b1.x = wp[16];     b1.y = wp[CCH + 16];
        b2.x = wp[32];     b2.y = wp[CCH + 32];
        b3.x = wp[48];     b3.y = wp[CCH + 48];
        acc0 = __builtin_amdgcn_wmma_f32_16x16x4_f32(false, a, false, b0, (short)0, acc0, false, false);
        acc1 = __builtin_amdgcn_wmma_f32_16x16x4_f32(false, a, false, b1, (short)0, acc1, false, false);
        acc2 = __builtin_amdgcn_wmma_f32_16x16x4_f32(false, a, false, b2, (short)0, acc2, false, false);
        acc3 = __builtin_amdgcn_wmma_f32_16x16x4_f32(false, a, false, b3, (short)0, acc3, false, false);
    }

    // D layout: VGPR v -> row = row0 + v + 8*halfSel, col = t*16 + l15
    v8f accs[4] = {acc0, acc1, acc2, acc3};
#pragma unroll
    for (int t = 0; t < 4; ++t) {
        const int col = t * 16 + l15;
        const float bb = bias[col];
        float* outp = h0 + (size_t)(row0 + halfSel * 8) * CCH + col;
#pragma unroll
        for (int v = 0; v < 8; ++v) {
            float val = accs[t][v] + bb;
            outp[(size_t)v * CCH] = fmaxf(val, 0.f);
        }
    }
#else
    // Scalar fallback (also keeps the host pass happy): same launch geometry.
    const int rowBase = blockIdx.x * 128;
    for (int idx = threadIdx.x; idx < 128 * CCH; idx += 256) {
        const int r = rowBase + (idx >> 6);
        if (r >= n) continue;
        const int c = idx & (CCH - 1);
        float accv = bias[c];
        const float* xr = x + (size_t)r * FIN;
        for (int k = 0; k < FIN; ++k) accv += xr[k] * W[(size_t)k * CCH + c];
        h0[(size_t)r * CCH + c] = fmaxf(accv, 0.f);
    }
#endif
}

// ---------------- propagation step: pure CSR row-gather SpMM -----------------
// One wave32 per node; lane owns channels {2*lane, 2*lane+1} (float2 = b64).
// Per edge per lane: one b64 record load (broadcast) + one b64 gather.
__global__ __launch_bounds__(256) void k_prop(float* __restrict__ hn,
                                              const float* __restrict__ hc,
                                              const float* __restrict__ x0,
                                              const float* __restrict__ d2s,
                                              const int* __restrict__ rs,
                                              const int* __restrict__ deg,
                                              const int2* __restrict__ cedge, int n) {
    const int lane = threadIdx.x & 31;
    const int node = blockIdx.x * (blockDim.x >> 5) + (threadIdx.x >> 5);
    if (node >= n) return;
    const int c2 = lane * 2;

    const float2 a = *(const float2*)(x0 + (size_t)node * CCH + c2);
    const float2 hs = *(const float2*)(hc + (size_t)node * CCH + c2);
    const float sl = d2s[node];
    float accx = ALPHA * a.x + sl * hs.x;
    float accy = ALPHA * a.y + sl * hs.y;

    const int start = rs[node];
    const int cnt = deg[node] - 1;  // in-edges excluding self-loop
    const int2* ep = cedge + start;
#pragma unroll 2
    for (int j = 0; j < cnt; ++j) {
        const int2 rec = ep[j];
        const float w = __int_as_float(rec.y);
        const float2 hv = *(const float2*)(hc + (size_t)rec.x * CCH + c2);
        accx += w * hv.x;
        accy += w * hv.y;
    }
    float2 r;
    r.x = accx;
    r.y = accy;
    *(float2*)(hn + (size_t)node * CCH + c2) = r;
}

// ---------------- row-wise log_softmax (one wave32 per row) ------------------
__global__ __launch_bounds__(256) void k_logsoftmax(const float* __restrict__ h,
                                                    float* __restrict__ out, int n) {
    const int lane = threadIdx.x & 31;
    const int row = blockIdx.x * (blockDim.x >> 5) + (threadIdx.x >> 5);
    if (row >= n) return;
    const float* p = h + (size_t)row * CCH;
    float v0 = p[lane], v1 = p[lane + 32];
    float m = fmaxf(v0, v1);
#pragma unroll
    for (int off = 16; off > 0; off >>= 1) m = fmaxf(m, __shfl_xor(m, off, 32));
    float s = expf(v0 - m) + expf(v1 - m);
#pragma unroll
    for (int off = 16; off > 0; off >>= 1) s += __shfl_xor(s, off, 32);
    const float lg = m + logf(s);
    float* o = out + (size_t)row * CCH;
    o[lane] = v0 - lg;
    o[lane + 32] = v1 - lg;
}

// ---------------------------------------------------------------------------
extern "C" void kernel_launch(void* const* d_in, const int* in_sizes, int n_in,
                              void* d_out, int out_size, void* d_ws, size_t ws_size,
                              hipStream_t stream) {
    const float* x = (const float*)d_in[0];
    const int* ei = (const int*)d_in[1];
    const float* W = (const float*)d_in[2];
    const float* bias = (const float*)d_in[3];
    float* out = (float*)d_out;

    const int N = in_sizes[0] / FIN;
    const int E = in_sizes[1] / 2;
    const int* src = ei;
    const int* dst = ei + E;
    const int nb = (N + SCAN_B - 1) / SCAN_B;  // 391 for N=100000 (<= SCAN_T2)

    // workspace carve (256B aligned)
    char* ws = (char*)d_ws;
    auto take = [&](size_t bytes) -> void* {
        void* p = (void*)ws;
        ws += (bytes + 255) & ~(size_t)255;
        return p;
    };
    float* buf0    = (float*)take((size_t)N * CCH * 4);  // x0 = h0
    float* buf1    = (float*)take((size_t)N * CCH * 4);
    float* buf2    = (float*)take((size_t)N * CCH * 4);
    float* dinv    = (float*)take((size_t)N * 4);
    float* d2s     = (float*)take((size_t)N * 4);
    int*   deg     = (int*)  take((size_t)N * 4);
    int*   rs      = (int*)  take((size_t)N * 4);   // CSR row starts (exclusive scan)
    int*   pos     = (int*)  take((size_t)N * 4);   // fill cursors
    int*   partial = (int*)  take((size_t)nb * 4);  // scan partials
    int2*  cedge   = (int2*) take((size_t)E * 8);   // CSR packed {src, w} records

    // degree + norms
    k_deg_init<<<(N + 255) / 256, 256, 0, stream>>>(deg, N);
    k_deg_count<<<2048, 256, 0, stream>>>(deg, dst, E);
    k_dinv<<<(N + 255) / 256, 256, 0, stream>>>(dinv, d2s, deg, N);

    // CSR build: scan + bucket fill
    k_scan1<<<nb, SCAN_B, 0, stream>>>(deg, rs, partial, N);
    k_scan2<<<1, SCAN_T2, 0, stream>>>(partial, nb);
    k_scan3<<<(N + 255) / 256, 256, 0, stream>>>(rs, pos, partial, N);
    k_csr_fill<<<2048, 256, 0, stream>>>(pos, cedge, src, dst, dinv, E);

    // h0 = relu(x @ W + b)
    k_gemm_relu<<<(N + 127) / 128, 256, 0, stream>>>(x, W, bias, buf0, N);

    // K propagation steps, ping-pong buf1/buf2, x0 stays in buf0
    const float* cur = buf0;
    float* nxt = buf1;
    const int propGrid = (N + 7) / 8;
    for (int it = 0; it < KSTEPS; ++it) {
        k_prop<<<propGrid, 256, 0, stream>>>(nxt, cur, buf0, d2s, rs, deg, cedge, N);
        const float* done = nxt;
        nxt = (it == 0) ? buf2 : (float*)cur;
        cur = done;
    }

    k_logsoftmax<<<(N + 7) / 8, 256, 0, stream>>>(cur, out, N);
}